// IISide_45595372814493
// MI455X (gfx1250) — compile-verified
//
#include <hip/hip_runtime.h>

typedef __attribute__((ext_vector_type(2))) float v2f;
typedef __attribute__((ext_vector_type(8))) float v8f;

constexpr int D = 64;   // embedding dim
constexpr int K = 10;   // knn neighbors per item

// ---------------- graph-norm SPMM path ----------------

__global__ void zero_f32(float* __restrict__ p, int n) {
    int i = blockIdx.x * blockDim.x + threadIdx.x;
    if (i < n) p[i] = 0.0f;
}

__global__ void deg_count(const int* __restrict__ rows, float* __restrict__ deg, int e) {
    int i = blockIdx.x * blockDim.x + threadIdx.x;
    if (i < e) atomicAdd(&deg[rows[i]], 1.0f);   // exact for small counts, order-independent
}

__global__ void deg_to_dis(float* __restrict__ d, int n) {
    int i = blockIdx.x * blockDim.x + threadIdx.x;
    if (i < n) d[i] = rsqrtf(d[i] + 1e-7f);      // deg^-1/2
}

// out[r,:] = dis[r] * sum_j dis[col_j] * x[col_j,:]   (edges of row r are [r*K, r*K+K))
__global__ void spmm_knn(const float* __restrict__ xin, const float* __restrict__ dis,
                         const int* __restrict__ cols, float* __restrict__ xout, int n) {
    int gid = blockIdx.x * blockDim.x + threadIdx.x;
    if (gid >= n * D) return;
    int row = gid >> 6;          // D == 64
    int d   = gid & (D - 1);
    const int base = row * K;
    float acc = 0.0f;
#pragma unroll
    for (int j = 0; j < K; ++j) {
        int c = cols[base + j];
        acc += dis[c] * xin[c * D + d];   // 64 consecutive lanes share c -> 256B coalesced, L2-hot
    }
    xout[gid] = dis[row] * acc;
}

// ---------------- WMMA f32 projector GEMM:  out[M,64] = X[M,KDIM] @ W[64,KDIM]^T + b ----------------
// Block: 128 threads = 4 waves; each wave owns a 16-col n-tile; block owns 16 rows.
// A staged through LDS in 16x128 chunks, pitch = 132 (== 4 mod 64 -> bank-conflict-free
// for the 16x4 f32 fragment pattern). B read directly (weights are L2/WGP$-resident).

template <int KDIM>
__global__ __launch_bounds__(128) void proj_gemm(const float* __restrict__ X,
                                                 const float* __restrict__ W,
                                                 const float* __restrict__ bias,
                                                 float* __restrict__ out, int n) {
    constexpr int CHUNK = 128;            // KDIM must be a multiple (1024, 384 both are)
    constexpr int PITCH = CHUNK + 4;
    __shared__ float As[16 * PITCH];

    const int tid   = threadIdx.x;
    const int lane  = tid & 31;
    const int wv    = tid >> 5;               // wave id -> n-tile
    const int m0    = blockIdx.x * 16;
    const int n0    = wv * 16;
    const int lrow  = lane & 15;              // fragment row (M for A, N for B)
    const int khalf = (lane >> 4) << 1;       // lanes 16-31 hold K+2..K+3

    const float* wrow = W + (size_t)(n0 + lrow) * KDIM + khalf;   // B^T row = weight row
    const float* arow = As + lrow * PITCH + khalf;

    v8f acc0 = {}; v8f acc1 = {}; v8f acc2 = {}; v8f acc3 = {};

    for (int kc = 0; kc < KDIM; kc += CHUNK) {
        // stage 16 x CHUNK slab of X into LDS (fully coalesced, uniform trip count)
#pragma unroll 4
        for (int i = tid; i < 16 * CHUNK; i += 128) {
            int r = i / CHUNK;
            int c = i - r * CHUNK;
            int gr = m0 + r; if (gr > n - 1) gr = n - 1;   // tail clamp (dup reads are fine)
            As[r * PITCH + c] = X[(size_t)gr * KDIM + kc + c];
        }
        __syncthreads();
        const float* wk = wrow + kc;
#pragma unroll
        for (int k0 = 0; k0 < CHUNK; k0 += 16) {
            v2f a0 = *(const v2f*)(arow + k0);
            v2f b0 = *(const v2f*)(wk + k0);
            acc0 = __builtin_amdgcn_wmma_f32_16x16x4_f32(false, a0, false, b0, (short)0, acc0, false, false);
            v2f a1 = *(const v2f*)(arow + k0 + 4);
            v2f b1 = *(const v2f*)(wk + k0 + 4);
            acc1 = __builtin_amdgcn_wmma_f32_16x16x4_f32(false, a1, false, b1, (short)0, acc1, false, false);
            v2f a2 = *(const v2f*)(arow + k0 + 8);
            v2f b2 = *(const v2f*)(wk + k0 + 8);
            acc2 = __builtin_amdgcn_wmma_f32_16x16x4_f32(false, a2, false, b2, (short)0, acc2, false, false);
            v2f a3 = *(const v2f*)(arow + k0 + 12);
            v2f b3 = *(const v2f*)(wk + k0 + 12);
            acc3 = __builtin_amdgcn_wmma_f32_16x16x4_f32(false, a3, false, b3, (short)0, acc3, false, false);
        }
        __syncthreads();
    }

    v8f acc = (acc0 + acc1) + (acc2 + acc3);
    // C/D layout: VGPR v -> M = v (+8 for lanes 16-31), N = n0 + (lane & 15)
    const int col   = n0 + lrow;
    const float bv  = bias[col];
    const int rbase = m0 + ((lane >> 4) << 3);
#pragma unroll
    for (int v = 0; v < 8; ++v) {
        int r = rbase + v;
        if (r < n) out[(size_t)r * D + col] = acc[v] + bv;
    }
}

// ---------------- driver ----------------

extern "C" void kernel_launch(void* const* d_in, const int* in_sizes, int n_in,
                              void* d_out, int out_size, void* d_ws, size_t ws_size,
                              hipStream_t stream) {
    const float* item  = (const float*)d_in[0];
    const float* vfeat = (const float*)d_in[1];
    const float* tfeat = (const float*)d_in[2];
    const float* vw    = (const float*)d_in[3];
    const float* vb    = (const float*)d_in[4];
    const float* tw    = (const float*)d_in[5];
    const float* tb    = (const float*)d_in[6];
    const int*   erows = (const int*)d_in[7];
    const int*   ecols = (const int*)d_in[8];
    float* out = (float*)d_out;

    const int n = in_sizes[0] / D;   // 100000
    const int e = in_sizes[8];       // 1,000,000

    float* ws  = (float*)d_ws;
    float* dis = ws;
    float* x1  = ws + (size_t)((n + 63) & ~63);
    float* x2  = x1 + (size_t)n * D;

    float* out_x = out;
    float* out_v = out + (size_t)n * D;
    float* out_t = out + 2 * (size_t)n * D;

    // edge weights (re-zeroed every call -> graph-replay safe)
    zero_f32 <<<(n + 255) / 256, 256, 0, stream>>>(dis, n);
    deg_count<<<(e + 255) / 256, 256, 0, stream>>>(erows, dis, e);
    deg_to_dis<<<(n + 255) / 256, 256, 0, stream>>>(dis, n);

    // 3 propagation layers (ping-pong, x fits in L2)
    const int nd = n * D;
    spmm_knn<<<(nd + 255) / 256, 256, 0, stream>>>(item, dis, ecols, x1, n);
    spmm_knn<<<(nd + 255) / 256, 256, 0, stream>>>(x1,   dis, ecols, x2, n);
    spmm_knn<<<(nd + 255) / 256, 256, 0, stream>>>(x2,   dis, ecols, out_x, n);

    // modality projectors via f32 WMMA
    const int mb = (n + 15) / 16;
    proj_gemm<1024><<<mb, 128, 0, stream>>>(vfeat, vw, vb, out_v, n);
    proj_gemm<384> <<<mb, 128, 0, stream>>>(tfeat, tw, tb, out_t, n);
}